// Model_32280974197209
// MI455X (gfx1250) — compile-verified
//
#include <hip/hip_runtime.h>
#include <stdint.h>

typedef __attribute__((ext_vector_type(2))) float v2f;
typedef __attribute__((ext_vector_type(8))) float v8f;

#define N_    4
#define CIN   128
#define HH    256
#define WW    256
#define GRP   8
#define ICG   16
#define OCG   16
#define OUTH  512
#define OUTW  512
#define KD    144      // contraction length = 16 ic * 3 * 3, permuted as (a*3+b)*16+ic
#define TILE_R 8
#define TILE_C 32
#define ROWS  10       // TILE_R + 2 halo
#define COLS  34       // TILE_C + 2 halo
#define ICP   18       // ic padded stride (even => 8B-aligned b64, conflict-free across 16 lanes)

// ---------------------------------------------------------------------------
// Kernel 1: fold upsample-FIR into the 3x3 conv -> 4 polyphase 3x3 kernels.
// Kbuf layout: [phase(4)][oc(128)][k'(144)], k' = (a*3+b)*16 + ic
// ---------------------------------------------------------------------------
__global__ void upconv_prep_kernel(const float* __restrict__ w,
                                   const float* __restrict__ f,
                                   float* __restrict__ Kbuf) {
  int idx = blockIdx.x * blockDim.x + threadIdx.x;
  if (idx >= 4 * CIN * KD) return;
  int k  = idx % KD;
  int oc = (idx / KD) % CIN;
  int ph = idx / (KD * CIN);
  int rh = ph >> 1, rw = ph & 1;
  int ab = k >> 4, ic = k & 15;
  int a = ab / 3, b = ab % 3;
  // even output rows use Keff taps {1,3,5}; odd rows use {0,2,4} (same for cols)
  int eh = rh ? (2 * a) : (2 * a + 1);
  int ew = rw ? (2 * b) : (2 * b + 1);
  // g = flipped filter * sqrt(gain)=2 per axis (gain = up*up = 4 total)
  float g[4] = {2.f * f[3], 2.f * f[2], 2.f * f[1], 2.f * f[0]};
  const float* wp = w + ((size_t)oc * ICG + ic) * 9;
  float acc = 0.f;
  for (int dh = 0; dh < 3; ++dh) {
    int s = eh - dh;
    if (s < 0 || s > 3) continue;
    for (int dw = 0; dw < 3; ++dw) {
      int t = ew - dw;
      if (t < 0 || t > 3) continue;
      acc += wp[dh * 3 + dw] * g[s] * g[t];
    }
  }
  Kbuf[((size_t)ph * CIN + oc) * KD + k] = acc;
}

// ---------------------------------------------------------------------------
// Kernel 2: implicit-GEMM polyphase conv with V_WMMA_F32_16X16X4_F32.
// Block = 256 threads = 8 waves. Each wave: one parity phase, 4 tile rows.
// x tile is DMA'd global->LDS with per-lane transposing gather via
// GLOBAL_LOAD_ASYNC_TO_LDS_B32 (ASYNCcnt path); A (16 oc x 144) resident in
// 72 VGPRs; B gathered from LDS with one ds_load_b64 + immediate per k-step.
// ---------------------------------------------------------------------------
__global__ __launch_bounds__(256, 1) void upconv_wmma_kernel(
    const float* __restrict__ x, const float* __restrict__ Kbuf,
    float* __restrict__ y) {
  __shared__ float xt[ROWS * COLS * ICP];   // 24480 B

  const int tid  = threadIdx.x;
  const int lane = tid & 31;
  const int wid  = tid >> 5;

  const int ng = blockIdx.z;                // n*8 + g
  const int n  = ng >> 3;
  const int g  = ng & 7;
  const int M0 = blockIdx.y * TILE_R;
  const int L0 = blockIdx.x * TILE_C;

  // ---- halo tile: async DMA gather global -> LDS (transpose to [row][col][ic])
  {
    const float* xb = x + (((size_t)n * CIN + g * ICG) * HH) * WW;
    for (int pair = wid; pair < ROWS * ICG; pair += 8) {
      int ic  = pair & 15;
      int rr  = pair >> 4;
      int row = M0 - 1 + rr;
      for (int c = lane; c < COLS; c += 32) {
        int col = L0 - 1 + c;
        float* dstp = &xt[(rr * COLS + c) * ICP + ic];
        if ((unsigned)row < HH && (unsigned)col < WW) {
          uint32_t ldsa = (uint32_t)(uintptr_t)dstp;   // LDS byte offset in low 32b
          uint64_t ga =
              (uint64_t)(uintptr_t)(xb + ((size_t)ic * HH + row) * WW + col);
          asm volatile("global_load_async_to_lds_b32 %0, %1, off"
                       :: "v"(ldsa), "v"(ga) : "memory");
        } else {
          *dstp = 0.f;   // zero halo
        }
      }
    }
  }

  // ---- per-wave phase + A operands into registers (overlaps with async DMA)
  const int ph    = wid & 3;         // parity phase
  const int rh    = ph >> 1, rw = ph & 1;
  const int rhalf = wid >> 2;        // which 4 rows of the 8-row tile
  const int half  = lane >> 4;       // WMMA K-pair select (k or k+2)
  const int m16   = lane & 15;       // A row (oc) / B col (pixel)

  v2f a[36];
  {
    const float* Arow =
        Kbuf + ((size_t)ph * CIN + g * OCG + m16) * KD + 2 * half;
#pragma unroll
    for (int kk = 0; kk < 36; ++kk)
      a[kk] = *(const v2f*)(Arow + 4 * kk);   // 8B-aligned (all offsets even)
  }

  // drain this wave's async-to-LDS transfers, then workgroup barrier
  asm volatile("s_wait_asynccnt 0x0" ::: "memory");
  __syncthreads();

#pragma unroll 1
  for (int r = 0; r < 4; ++r) {
    const int mrow = rhalf * 4 + r;           // tile-local input row
#pragma unroll 1
    for (int cg = 0; cg < 2; ++cg) {
      const int lcol = cg * 16 + m16;         // tile-local input col (per lane)
      // x row = (M0+mrow)-1+a -> tile index mrow+a; col likewise lcol+b
      const float* Bb = xt + (mrow * COLS + lcol) * ICP + 2 * half;
      v8f acc = {};
#pragma unroll
      for (int kk = 0; kk < 36; ++kk) {
        const int ab   = kk >> 2;             // compile-time: (a,b) tap
        const int aoff = ab / 3;
        const int boff = ab % 3;
        const int ico  = (kk & 3) * 4;        // compile-time ic base (even)
        v2f bv = *(const v2f*)(Bb + (aoff * COLS + boff) * ICP + ico);
        acc = __builtin_amdgcn_wmma_f32_16x16x4_f32(
            false, a[kk], false, bv, (short)0, acc, false, false);
      }
      const int p = 2 * (M0 + mrow) + rh;
      const int q = 2 * (L0 + lcol) + rw;
      float* dst = y + ((size_t)(n * CIN) * OUTH + p) * OUTW + q;
#pragma unroll
      for (int v = 0; v < 8; ++v) {
        int oc = g * OCG + 8 * half + v;      // D row = v + 8*(lane/16)
        dst[(size_t)oc * OUTH * OUTW] = acc[v];
      }
    }
  }
}

// ---------------------------------------------------------------------------
extern "C" void kernel_launch(void* const* d_in, const int* in_sizes, int n_in,
                              void* d_out, int out_size, void* d_ws, size_t ws_size,
                              hipStream_t stream) {
  const float* x = (const float*)d_in[0];
  const float* f = (const float*)d_in[1];
  const float* w = (const float*)d_in[2];
  float* y    = (float*)d_out;
  float* Kbuf = (float*)d_ws;     // 4*128*144*4 = 294912 B

  int total = 4 * CIN * KD;
  upconv_prep_kernel<<<(total + 255) / 256, 256, 0, stream>>>(w, f, Kbuf);

  dim3 grid(WW / TILE_C, HH / TILE_R, N_ * GRP);   // (8, 32, 32)
  upconv_wmma_kernel<<<grid, 256, 0, stream>>>(x, Kbuf, y);
}